// Decoder_65704409694293
// MI455X (gfx1250) — compile-verified
//
#include <hip/hip_runtime.h>
#include <hip/hip_bf16.h>

typedef float v2f __attribute__((ext_vector_type(2)));
typedef float v8f __attribute__((ext_vector_type(8)));

__device__ __forceinline__ void atomAddF(float* p, float v) {
    // hardware global_atomic_add_f32 (no CAS loop)
    unsafeAtomicAdd(p, v);
}

// ---------------- degree / norm precompute ----------------

__global__ void fill_deg_k(float* __restrict__ deg, int n) {
    int i = blockIdx.x * blockDim.x + threadIdx.x;
    if (i < n) deg[i] = 1.0f;  // self-loop contribution
}

__global__ void accum_deg_k(float* __restrict__ deg, const int* __restrict__ dst, int e) {
    int i = blockIdx.x * blockDim.x + threadIdx.x;
    if (i < e) atomAddF(&deg[dst[i]], 1.0f);
}

__global__ void rsqrt_k(float* __restrict__ deg, int n) {
    int i = blockIdx.x * blockDim.x + threadIdx.x;
    if (i < n) deg[i] = rsqrtf(deg[i]);  // deg >= 1 always (self loop)
}

__global__ void norm_k(float* __restrict__ norm, const float* __restrict__ dinv,
                       const int* __restrict__ src, const int* __restrict__ dst, int e) {
    int i = blockIdx.x * blockDim.x + threadIdx.x;
    if (i < e) norm[i] = dinv[src[i]] * dinv[dst[i]];
}

// ---------------- dense transform: H = (relu?)(A) @ W via V_WMMA_F32_16X16X4_F32 ----
// One wave per 16-row M-tile. A fragment loaded once (K/4 x v2f via b64 loads),
// reused across all F/16 N-tiles. Fully unrolled: K/4 * F/16 wmma ops per wave.
// Requires full tiles (rows handled here are < tilesM*16); remainder rows go to
// the scalar kernel below (zero blocks when n % 16 == 0).

template <int K, int F, bool RELU>
__global__ void gemm_wmma_k(const float* __restrict__ A, const float* __restrict__ W,
                            float* __restrict__ H, int tilesM) {
    constexpr int KF = K / 4;
    int wave = blockIdx.x * (blockDim.x >> 5) + (threadIdx.x >> 5);
    if (wave >= tilesM) return;                 // wave-uniform: EXEC stays full

    int lane = threadIdx.x & 31;
    int l16  = lane & 15;
    int half = lane >> 4;                       // 0: K={0,1}, 1: K={2,3} per 4-group

    // Preload A fragments: lane l16 = row (wave*16 + l16), K pair at i*4 + half*2
    const float* arow = A + ((size_t)wave * 16 + l16) * K;
    v2f afr[KF];
#pragma unroll
    for (int i = 0; i < KF; ++i) {
        v2f a = *(const v2f*)(arow + i * 4 + half * 2);   // 8B-aligned (even kb)
        if (RELU) { a.x = fmaxf(a.x, 0.f); a.y = fmaxf(a.y, 0.f); }
        afr[i] = a;
    }

    for (int tn = 0; tn < F / 16; ++tn) {
        int n = tn * 16 + l16;
        v8f acc = {0.f, 0.f, 0.f, 0.f, 0.f, 0.f, 0.f, 0.f};
#pragma unroll
        for (int i = 0; i < KF; ++i) {
            int kb = i * 4 + half * 2;
            v2f b;
            b.x = W[kb * F + n];
            b.y = W[(kb + 1) * F + n];
            acc = __builtin_amdgcn_wmma_f32_16x16x4_f32(
                false, afr[i], false, b, (short)0, acc, false, false);
        }
        // D: VGPR r -> row (wave*16 + half*8 + r), col n
        float* hp = H + ((size_t)wave * 16 + half * 8) * F + n;
#pragma unroll
        for (int r = 0; r < 8; ++r) hp[r * F] = acc[r];
    }
}

// Scalar remainder for rows [row0, n) — launches zero blocks when n % 16 == 0.
__global__ void gemm_rem_k(const float* __restrict__ A, const float* __restrict__ W,
                           float* __restrict__ H, int row0, int n, int K, int F, int relu) {
    int t = blockIdx.x * blockDim.x + threadIdx.x;
    int rows = n - row0;
    if (t >= rows * F) return;
    int row = row0 + t / F;
    int col = t % F;
    float s = 0.f;
    for (int k = 0; k < K; ++k) {
        float a = A[(size_t)row * K + k];
        if (relu) a = fmaxf(a, 0.f);
        s += a * W[k * F + col];
    }
    H[(size_t)row * F + col] = s;
}

// ---------------- agg init: agg = bias + dinv^2 * h  (self loop folded in) ---------

template <int F>
__global__ void initagg_k(const float* __restrict__ H, const float* __restrict__ bias,
                          const float* __restrict__ dinv, float* __restrict__ agg,
                          int nNodes) {
    constexpr int F4 = F / 4;
    int tid = blockIdx.x * blockDim.x + threadIdx.x;
    if (tid >= nNodes * F4) return;
    int nIdx = tid / F4;
    int q    = tid % F4;
    float di = dinv[nIdx];
    float s  = di * di;
    float4 v = ((const float4*)H)[tid];
    float4 b = ((const float4*)bias)[q];
    float4 o;
    o.x = b.x + s * v.x; o.y = b.y + s * v.y;
    o.z = b.z + s * v.z; o.w = b.w + s * v.w;
    ((float4*)agg)[tid] = o;
}

// ---------------- edge scatter: agg[dst] += norm * h[src] --------------------------

template <int F>
__global__ void scatter_k(const float* __restrict__ H, const float* __restrict__ norm,
                          const int* __restrict__ src, const int* __restrict__ dst,
                          float* __restrict__ agg, int e) {
    constexpr int F4 = F / 4;
    int tid = blockIdx.x * blockDim.x + threadIdx.x;
    if (tid >= e * F4) return;
    int ei = tid / F4;
    int q  = tid % F4;
    int s  = src[ei];
    int d  = dst[ei];
    float w = norm[ei];
    float4 v = *((const float4*)(H + (size_t)s * F) + q);
    float* ap = agg + (size_t)d * F + q * 4;
    atomAddF(ap + 0, w * v.x);
    atomAddF(ap + 1, w * v.y);
    atomAddF(ap + 2, w * v.z);
    atomAddF(ap + 3, w * v.w);
}

// ---------------- host orchestration ----------------------------------------------

template <int K, int F, bool RELU>
static void launch_layer(const float* in, const float* W, const float* bias,
                         float* h, float* agg,
                         const int* src, const int* dst, const float* norm,
                         const float* dinv, int n, int e, hipStream_t stream) {
    int tilesM = n / 16;
    int rem    = n % 16;
    if (tilesM > 0) {
        int gemmBlocks = (tilesM + 3) / 4;  // 4 waves of 32 per block
        gemm_wmma_k<K, F, RELU><<<gemmBlocks, 128, 0, stream>>>(in, W, h, tilesM);
    }
    if (rem > 0) {
        int t = rem * F;
        gemm_rem_k<<<(t + 255) / 256, 256, 0, stream>>>(in, W, h, tilesM * 16, n, K, F,
                                                        RELU ? 1 : 0);
    }
    int tot1 = n * (F / 4);
    initagg_k<F><<<(tot1 + 255) / 256, 256, 0, stream>>>(h, bias, dinv, agg, n);
    int tot2 = e * (F / 4);
    scatter_k<F><<<(tot2 + 255) / 256, 256, 0, stream>>>(h, norm, src, dst, agg, e);
}

extern "C" void kernel_launch(void* const* d_in, const int* in_sizes, int n_in,
                              void* d_out, int out_size, void* d_ws, size_t ws_size,
                              hipStream_t stream) {
    const float* x  = (const float*)d_in[0];
    const int*   ei = (const int*)d_in[1];
    const float* W1 = (const float*)d_in[2];
    const float* b1 = (const float*)d_in[3];
    const float* W2 = (const float*)d_in[4];
    const float* b2 = (const float*)d_in[5];
    const float* W3 = (const float*)d_in[6];
    const float* b3 = (const float*)d_in[7];

    const int n = in_sizes[0] / 16;   // 100000
    const int e = in_sizes[1] / 2;    // 1600000
    const int* src = ei;
    const int* dst = ei + e;

    float* wsf    = (float*)d_ws;
    float* dinv   = wsf;                      // N floats (deg -> dinv in place)
    float* norm   = dinv + n;                 // E floats
    float* hbuf   = norm + e;                 // N*128 floats (reused per layer)
    float* aggbuf = hbuf + (size_t)n * 128;   // N*64 floats (agg1/agg2, reused)
    float* out    = (float*)d_out;            // N*128 (agg3)

    // --- cached symmetric normalization ---
    fill_deg_k <<<(n + 255) / 256, 256, 0, stream>>>(dinv, n);
    accum_deg_k<<<(e + 255) / 256, 256, 0, stream>>>(dinv, dst, e);
    rsqrt_k    <<<(n + 255) / 256, 256, 0, stream>>>(dinv, n);
    norm_k     <<<(e + 255) / 256, 256, 0, stream>>>(norm, dinv, src, dst, e);

    // --- layer 1: 16 -> 32, input x (no relu on input) ---
    launch_layer<16, 32, false>(x, W1, b1, hbuf, aggbuf, src, dst, norm, dinv, n, e, stream);
    // --- layer 2: 32 -> 64, input agg1 (relu folded into GEMM A-load) ---
    launch_layer<32, 64, true>(aggbuf, W2, b2, hbuf, aggbuf, src, dst, norm, dinv, n, e, stream);
    // --- layer 3: 64 -> 128, input agg2, output directly to d_out (no final relu) ---
    launch_layer<64, 128, true>(aggbuf, W3, b3, hbuf, out, src, dst, norm, dinv, n, e, stream);
}